// JacobiConv_2310692405499
// MI455X (gfx1250) — compile-verified
//
#include <hip/hip_runtime.h>
#include <cstdint>
#include <cstddef>

#define NN 100000
#define NE 3200000
#define DF 128
#define KORD 10
#define DEPTH 4            // async gather pipeline depth (power of 2)

// ---------------------------------------------------------------- utilities
__global__ __launch_bounds__(256) void k_zero(float* __restrict__ p, int n) {
    int i = blockIdx.x * 256 + threadIdx.x;
    if (i < n) p[i] = 0.0f;
}

__global__ __launch_bounds__(256) void k_deg(const int* __restrict__ row,
                                             float* __restrict__ deg, int nE) {
    int i = blockIdx.x * 256 + threadIdx.x;
    if (i < nE) atomicAdd(&deg[row[i]], 1.0f);
}

__global__ __launch_bounds__(256) void k_rsqrt(float* __restrict__ p, int n) {
    int i = blockIdx.x * 256 + threadIdx.x;
    if (i < n) {
        float d = p[i];
        p[i] = (d > 0.0f) ? rsqrtf(d) : 0.0f;
    }
}

// out = alpha[0]*x ; P1init = p1s*x        (float4-vectorized over N*D)
__global__ __launch_bounds__(256) void k_init(const float4* __restrict__ x,
                                              float4* __restrict__ out,
                                              float4* __restrict__ p1,
                                              const float* __restrict__ alpha,
                                              float p1s, int n4) {
    int i = blockIdx.x * 256 + threadIdx.x;
    if (i >= n4) return;
    float a0 = alpha[0];
    float4 xv = x[i];
    out[i] = make_float4(a0 * xv.x, a0 * xv.y, a0 * xv.z, a0 * xv.w);
    p1[i]  = make_float4(p1s * xv.x, p1s * xv.y, p1s * xv.z, p1s * xv.w);
}

// Pnew = cp1*P1 + cp2*P2 ; out += beta[kb]*prod(gamma[0..kb-1]) * P1
// NOTE: pn may alias p2 (in-place elementwise) -> no restrict on those.
__global__ __launch_bounds__(256) void k_combine(const float4* __restrict__ p1,
                                                 const float4* p2,
                                                 float4* pn,
                                                 float4* __restrict__ out,
                                                 const float* __restrict__ beta,
                                                 const float* __restrict__ gamma,
                                                 int kb, float cp1, float cp2, int n4) {
    int i = blockIdx.x * 256 + threadIdx.x;
    if (i >= n4) return;
    float gp = 1.0f;
    for (int j = 0; j < kb; ++j) gp *= gamma[j];
    float s = beta[kb] * gp;
    float4 a = p1[i];
    float4 b = p2[i];
    float4 o = out[i];
    o.x += s * a.x; o.y += s * a.y; o.z += s * a.z; o.w += s * a.w;
    out[i] = o;
    pn[i] = make_float4(cp1 * a.x + cp2 * b.x, cp1 * a.y + cp2 * b.y,
                        cp1 * a.z + cp2 * b.z, cp1 * a.w + cp2 * b.w);
}

// out += beta[kb]*prod(gamma[0..kb-1]) * P
__global__ __launch_bounds__(256) void k_accum(const float4* __restrict__ p,
                                               float4* __restrict__ out,
                                               const float* __restrict__ beta,
                                               const float* __restrict__ gamma,
                                               int kb, int n4) {
    int i = blockIdx.x * 256 + threadIdx.x;
    if (i >= n4) return;
    float gp = 1.0f;
    for (int j = 0; j < kb; ++j) gp *= gamma[j];
    float s = beta[kb] * gp;
    float4 a = p[i];
    float4 o = out[i];
    o.x += s * a.x; o.y += s * a.y; o.z += s * a.z; o.w += s * a.w;
    out[i] = o;
}

// -------------------------------------------------- async-pipelined scatter
// dst[row[e]][:] += scale * dis[row]*dis[col] * src[col[e]][:]
// 1 wave per edge: lane l covers features [4l,4l+4). A DEPTH-deep ring of
// GLOBAL_LOAD_ASYNC_TO_LDS_B128 gathers keeps DEPTH edges' 512B rows in
// flight per wave (ASYNCcnt pipeline); edge indices are wave-uniform scalar
// loads, so they are simply re-read at consume time instead of queued.
__global__ __launch_bounds__(256) void k_scatter(const float* __restrict__ src,
                                                 float* __restrict__ dst,
                                                 const int* __restrict__ row,
                                                 const int* __restrict__ col,
                                                 const float* __restrict__ dis,
                                                 float scale, int nE, int nWaves) {
    __shared__ float sh[DEPTH][8][DF];         // 16 KB: 4 slots x 8 waves x 128 f32
    const int lane = threadIdx.x & 31;
    const int w    = threadIdx.x >> 5;
    int e = blockIdx.x * 8 + w;                // this wave's first edge

    // prologue: fill the pipeline
    #pragma unroll
    for (int i = 0; i < DEPTH; ++i) {
        int ei = e + i * nWaves;
        if (ei < nE) {
            int c = col[ei];
            const float* gp = src + (size_t)c * DF + lane * 4;
            uint32_t la = (uint32_t)(uintptr_t)&sh[i][w][lane * 4];
            asm volatile("global_load_async_to_lds_b128 %0, %1, off"
                         :: "v"(la), "v"(gp) : "memory");
        }
    }

    int slot = 0;
    while (e < nE) {
        // steady state: DEPTH in flight -> wait for the oldest only
        if (e + (DEPTH - 1) * nWaves < nE) {
            asm volatile("s_wait_asynccnt 3" ::: "memory");
        } else {
            asm volatile("s_wait_asynccnt 0" ::: "memory");   // pipeline drain (tail)
        }
        int r = row[e];
        int c = col[e];
        float wt = scale * dis[r] * dis[c];
        float4 v = *(const float4*)&sh[slot][w][lane * 4];
        float* dp = dst + (size_t)r * DF + lane * 4;
        atomicAdd(dp + 0, wt * v.x);
        atomicAdd(dp + 1, wt * v.y);
        atomicAdd(dp + 2, wt * v.z);
        atomicAdd(dp + 3, wt * v.w);
        // refill the just-freed slot (ds_load completion already enforced
        // before the atomics consumed v, so no LDS WAR hazard)
        int eN = e + DEPTH * nWaves;
        if (eN < nE) {
            int cN = col[eN];
            const float* gp = src + (size_t)cN * DF + lane * 4;
            uint32_t la = (uint32_t)(uintptr_t)&sh[slot][w][lane * 4];
            asm volatile("global_load_async_to_lds_b128 %0, %1, off"
                         :: "v"(la), "v"(gp) : "memory");
        }
        e += nWaves;
        slot = (slot + 1) & (DEPTH - 1);
    }
}

// ------------------------------------------------------------------- driver
extern "C" void kernel_launch(void* const* d_in, const int* in_sizes, int n_in,
                              void* d_out, int out_size, void* d_ws, size_t ws_size,
                              hipStream_t stream) {
    const float* x     = (const float*)d_in[0];   // [NN, DF]
    const float* alpha = (const float*)d_in[1];   // [K+1]
    const float* beta  = (const float*)d_in[2];   // [K+1]
    const float* gamma = (const float*)d_in[3];   // [K]
    const int*   eidx  = (const int*)d_in[4];     // [2, NE] row-major
    const int*   row   = eidx;
    const int*   col   = eidx + NE;
    float* out = (float*)d_out;

    float* ws  = (float*)d_ws;
    float* dis = ws;                                    // NN floats
    float* PA  = ws + 100352;                           // NN rounded up to 1024
    float* PB  = PA + (size_t)NN * DF;                  // second N*D buffer

    const int n4 = NN * DF / 4;                         // 3,200,000 float4 elems
    const dim3 B(256);
    const dim3 G_nd((n4 + 255) / 256);
    const dim3 G_n((NN + 255) / 256);
    const dim3 G_e((NE + 255) / 256);
    const int  SC_BLOCKS = 4096;                        // 32768 waves
    const int  SC_WAVES  = SC_BLOCKS * 8;

    // host-side Jacobi coefficients (pure functions of k,a,b,lmax)
    const double a = 0.5, b = 0.5, lmax = 2.0;
    const double cL = 2.0 / lmax;                       // == 1.0
    const float p1_init = (float)((a - b) / 2.0 + (a + b + 2.0) / 2.0 * (cL - 1.0));
    const float p1_scat = (float)(-(a + b + 2.0) / 2.0 * cL);

    // degree -> dis = rsqrt(deg)
    k_zero<<<G_n, B, 0, stream>>>(dis, NN);
    k_deg<<<G_e, B, 0, stream>>>(row, dis, NE);
    k_rsqrt<<<G_n, B, 0, stream>>>(dis, NN);

    // P1 (into PA) and out = alpha0*x
    k_init<<<G_nd, B, 0, stream>>>((const float4*)x, (float4*)out, (float4*)PA,
                                   alpha, p1_init, n4);
    k_scatter<<<SC_BLOCKS, B, 0, stream>>>(x, PA, row, col, dis, p1_scat, NE, SC_WAVES);

    // recurrence: ping-pong PA/PB; Pnew reuses the retired P_{k-2} buffer
    const float* p2 = x;
    float* p1 = PA;
    int newIdx = 1;
    float* bufs[2] = {PA, PB};
    for (int k = 2; k <= KORD; ++k) {
        const double ab = a + b;
        const double th   = (2.0 * k + ab) * (2.0 * k + ab - 1.0) / (2.0 * k * (k + ab));
        const double thp  = (2.0 * k + ab - 1.0) * (a * a - b * b) /
                            (2.0 * k * (k + ab) * (2.0 * k + ab - 2.0));
        const double thpp = (k + a - 1.0) * (k + b - 1.0) * (2.0 * k + ab) /
                            ((double)k * (k + ab) * (2.0 * k + ab - 2.0));
        const float cp1  = (float)(th * (cL - 1.0) + thp);
        const float cp2  = (float)(-thpp);
        const float scat = (float)(-th * cL);

        float* pn = bufs[newIdx];
        // fold out += beta[k-1]*Gamma_{k-1}*P_{k-1} into the Pk-init pass
        k_combine<<<G_nd, B, 0, stream>>>((const float4*)p1, (const float4*)p2,
                                          (float4*)pn, (float4*)out,
                                          beta, gamma, k - 1, cp1, cp2, n4);
        k_scatter<<<SC_BLOCKS, B, 0, stream>>>(p1, pn, row, col, dis, scat, NE, SC_WAVES);

        p2 = p1; p1 = pn; newIdx ^= 1;
    }

    // final out += beta[K]*Gamma_K*P_K
    k_accum<<<G_nd, B, 0, stream>>>((const float4*)p1, (float4*)out,
                                    beta, gamma, KORD, n4);
}